// SPCSampler_13142599926288
// MI455X (gfx1250) — compile-verified
//
#include <hip/hip_runtime.h>
#include <math.h>

// ---------------- problem constants (fixed by the reference) ----------------
#define BATCH   2
#define NPTS    65536
#define NROI    128
#define KKP     2048         // NUM_KEYPOINTS
#define NSEC    6
#define PI_F    3.14159265358979323846f
#define POS_F   1e10f

typedef __attribute__((ext_vector_type(2))) float v2f;
typedef __attribute__((ext_vector_type(8))) float v8f;

// ============================================================================
// Kernel 1: ROI prep.  centers4 = (cx, cy, cz+dz/2, |c|^2),  T = |dims/2|+1.6
// Also zeroes the per-batch valid counters (ws is not re-poisoned by harness).
// ============================================================================
__global__ void spc_prep(const float* __restrict__ rois,
                         float4* __restrict__ centers4,
                         float*  __restrict__ Tarr,
                         int*    __restrict__ validCount) {
    int t = threadIdx.x;
    if (t < BATCH) validCount[t] = 0;
    if (t < BATCH * NROI) {
        const float* r = rois + (size_t)t * 7;
        float cx = r[0], cy = r[1], cz = r[2] + 0.5f * r[5];
        float hx = 0.5f * r[3], hy = 0.5f * r[4], hz = 0.5f * r[5];
        centers4[t] = make_float4(cx, cy, cz, cx*cx + cy*cy + cz*cz);
        Tarr[t] = sqrtf(hx*hx + hy*hy + hz*hz) + 1.6f;   // SAMPLE_RADIUS
    }
}

// ============================================================================
// Kernel 2: classification via V_WMMA_F32_16X16X4_F32.
// One wave handles 16 points vs all 128 ROIs (8 WMMA tiles).
//   A (16x4, rows = ROIs):  K0=cx  K1=cy  K2=cz  K3=|c|^2
//   B (4x16, cols = points):K0=-2x K1=-2y K2=-2z K3=1
//   => D[m,p] = |c_m|^2 - 2 p.c_m ;  d^2 = |p|^2 + D[m,p]
// ISA wave32 layouts: A VGPR0={K0|K2}, VGPR1={K1|K3} (lane halves);
// D: lane = column(point), VGPR v = row m (lanes16-31: m=v+8).
// Writes svalid[b][n] = sector(0..6) if valid else -1; counts valid per batch.
// ============================================================================
__global__ void spc_classify(const float*  __restrict__ points,
                             const float4* __restrict__ centers4,
                             const float*  __restrict__ Tarr,
                             int* __restrict__ svalid,
                             int* __restrict__ validCount) {
    const int lane = threadIdx.x & 31;
    const int wave = threadIdx.x >> 5;
    const int blocksPerB = NPTS / (16 * 8);            // 512
    const int b = blockIdx.x / blocksPerB;
    const int blkInB = blockIdx.x % blocksPerB;
    const int tileBase = (blkInB * 8 + wave) * 16;
    const int p = tileBase + (lane & 15);
    const bool hi = lane >= 16;

    const float* pp = points + ((size_t)b * NPTS + p) * 3;
    const float px = pp[0], py = pp[1], pz = pp[2];
    const float pn2 = px*px + py*py + pz*pz;

    v2f Bm;
    Bm[0] = hi ? (-2.0f * pz) : (-2.0f * px);          // K2 | K0
    Bm[1] = hi ? 1.0f         : (-2.0f * py);          // K3 | K1

    float bestD = 3.4e38f;
    int   bestM = 0;
    const float4* cb = centers4 + b * NROI;
    #pragma unroll
    for (int t = 0; t < NROI / 16; ++t) {
        float4 c = cb[t * 16 + (lane & 15)];
        v2f Am;
        Am[0] = hi ? c.z : c.x;                        // K2 | K0
        Am[1] = hi ? c.w : c.y;                        // K3 | K1
        v8f acc = {};
        acc = __builtin_amdgcn_wmma_f32_16x16x4_f32(
            false, Am, false, Bm, (short)0, acc, false, false);
        #pragma unroll
        for (int v = 0; v < 8; ++v) {
            float d2 = pn2 + acc[v];
            int m = t * 16 + (hi ? (8 + v) : v);
            if (d2 < bestD) { bestD = d2; bestM = m; } // ascending m -> first-min
        }
    }
    // combine lane pair (p, p+16); prefer lower ROI index on tie (jnp.argmin)
    float oD = __shfl_xor(bestD, 16, 32);
    int   oM = __shfl_xor(bestM, 16, 32);
    if (oD < bestD || (oD == bestD && oM < bestM)) { bestD = oD; bestM = oM; }

    float T = Tarr[b * NROI + bestM];
    bool valid = bestD < T * T;                        // min_dis < rmax+radius

    float ang = atan2f(py, px) + PI_F;                 // [0, 2pi]
    int sec = (int)floorf(ang * ((float)NSEC / (2.0f * PI_F)));
    if (sec < 0) sec = 0;
    if (sec > NSEC) sec = NSEC;                        // clip(...,0,NUM_SECTORS)

    if (!hi) svalid[(size_t)b * NPTS + p] = valid ? sec : -1;
    unsigned long long bal = __ballot((valid && !hi) ? 1 : 0);
    if (lane == 0) atomicAdd(&validCount[b], __popcll(bal));
}

// ============================================================================
// Kernel 3: order-preserving compaction per (batch, sector) using an LDS scan.
// compact[bs][pos] = (x, y, z, bitcast(original index)), cnt[bs] = count.
// ============================================================================
__global__ __launch_bounds__(1024)
void spc_compact(const float* __restrict__ points,
                 const int*   __restrict__ svalid,
                 float4* __restrict__ compactBuf,
                 int*    __restrict__ cnt) {
    __shared__ int scanb[1024];
    const int bs = blockIdx.x;
    const int b = bs / NSEC, s = bs % NSEC;
    const int tid = threadIdx.x;
    const int* sv = svalid + (size_t)b * NPTS;
    float4* outp = compactBuf + (size_t)bs * NPTS;
    int base = 0;
    for (int c0 = 0; c0 < NPTS; c0 += 1024) {
        int n = c0 + tid;
        int flag = (sv[n] == s) ? 1 : 0;
        scanb[tid] = flag;
        __syncthreads();
        #pragma unroll
        for (int off = 1; off < 1024; off <<= 1) {     // Hillis-Steele inclusive
            int v = (tid >= off) ? scanb[tid - off] : 0;
            __syncthreads();
            scanb[tid] += v;
            __syncthreads();
        }
        if (flag) {
            int pos = base + scanb[tid] - 1;
            const float* pp = points + ((size_t)b * NPTS + n) * 3;
            outp[pos] = make_float4(pp[0], pp[1], pp[2], __int_as_float(n));
        }
        base += scanb[1023];
        __syncthreads();
    }
    if (tid == 0) cnt[bs] = base;
}

// ============================================================================
// Kernel 4: per-sector budgets n_k = min(cnt, ceil(cnt*K/total)), offsets,
// s_total = max(sum n_k, 1).  Tiny.
// ============================================================================
__global__ void spc_budget(const int* __restrict__ validCount,
                           const int* __restrict__ cnt,
                           int* __restrict__ nk, int* __restrict__ offs,
                           int* __restrict__ stotal) {
    int b = threadIdx.x;
    if (b >= BATCH) return;
    int total = validCount[b] > 1 ? validCount[b] : 1;
    int acc = 0;
    for (int s = 0; s < NSEC; ++s) {
        int c = cnt[b * NSEC + s];
        int q = (int)ceilf(((float)c * (float)KKP) / (float)total);
        int v = c < q ? c : q;
        nk[b * NSEC + s] = v;
        offs[b * NSEC + s] = acc;
        acc += v;
    }
    stotal[b] = acc > 0 ? acc : 1;
}

// ============================================================================
// Kernel 5: masked FPS per (batch, sector); dmin lives in LDS (<=256KB).
// Only n_k-1 sequential iterations (reference keeps only the first n_k picks).
// Block = 1024 threads = 32 waves; argmax with first-index tie-break.
// ============================================================================
__global__ __launch_bounds__(1024)
void spc_fps(const float4* __restrict__ compactBuf,
             const int* __restrict__ cnt, const int* __restrict__ nk,
             int* __restrict__ fpsOut) {
    extern __shared__ char smem[];
    float* redV = (float*)smem;                        // [33]
    int*   redI = (int*)(smem + 33 * sizeof(float));   // [33]
    float* dmin = (float*)(smem + 272);                // [<=NPTS]

    const int bs = blockIdx.x;
    const int count = cnt[bs];
    const int n_k = nk[bs];
    if (n_k <= 0) return;                              // uniform exit
    const float4* pts = compactBuf + (size_t)bs * NPTS;
    int* out = fpsOut + (size_t)bs * KKP;
    const int tid = threadIdx.x;
    const int lane = tid & 31, w = tid >> 5;

    for (int i = tid; i < count; i += 1024) dmin[i] = POS_F;
    if (tid == 0) out[0] = __float_as_int(pts[0].w);   // first valid point
    __syncthreads();

    int last = 0;
    for (int k = 1; k < n_k; ++k) {
        float4 lp = pts[last];                         // uniform broadcast load
        float bv = -3.4e38f;
        int   bi = 0x7fffffff;
        for (int i = tid; i < count; i += 1024) {
            __builtin_prefetch(pts + i + 1024, 0, 1);  // global_prefetch_b8
            float4 q = pts[i];
            float dx = q.x - lp.x, dy = q.y - lp.y, dz = q.z - lp.z;
            float d = dx*dx + dy*dy + dz*dz;
            float dm = dmin[i];
            dm = d < dm ? d : dm;
            dmin[i] = dm;
            if (dm > bv) { bv = dm; bi = i; }          // ascending i -> first-max
        }
        #pragma unroll
        for (int off = 16; off >= 1; off >>= 1) {      // intra-wave (val,idx) max
            float ov = __shfl_xor(bv, off, 32);
            int   oi = __shfl_xor(bi, off, 32);
            if (ov > bv || (ov == bv && oi < bi)) { bv = ov; bi = oi; }
        }
        if (lane == 0) { redV[w] = bv; redI[w] = bi; }
        __syncthreads();
        if (w == 0) {
            bv = redV[lane]; bi = redI[lane];
            #pragma unroll
            for (int off = 16; off >= 1; off >>= 1) {
                float ov = __shfl_xor(bv, off, 32);
                int   oi = __shfl_xor(bi, off, 32);
                if (ov > bv || (ov == bv && oi < bi)) { bv = ov; bi = oi; }
            }
            if (lane == 0) { redV[32] = bv; redI[32] = bi; }
        }
        __syncthreads();
        last = redI[32];
        if (tid == 0) out[k] = __float_as_int(pts[last].w);
    }
}

// ============================================================================
// Kernel 6: assemble output: slot = i % s_total -> (sector, j) -> gather xyz.
// Uncovered slots fall back to point 0 (matches zeros-initialized buf).
// ============================================================================
__global__ void spc_assemble(const float* __restrict__ points,
                             const int* __restrict__ nk,
                             const int* __restrict__ offs,
                             const int* __restrict__ stotal,
                             const int* __restrict__ fpsOut,
                             float* __restrict__ out) {
    int gi = blockIdx.x * blockDim.x + threadIdx.x;
    if (gi >= BATCH * KKP) return;
    int b = gi / KKP;
    int j = gi % KKP;
    int st = stotal[b];
    int slot = j % st;
    int idx = 0;
    #pragma unroll
    for (int s = 0; s < NSEC; ++s) {
        int o = offs[b * NSEC + s];
        int v = nk[b * NSEC + s];
        if (slot >= o && slot < o + v) {
            idx = fpsOut[((size_t)(b * NSEC + s)) * KKP + (slot - o)];
            break;
        }
    }
    const float* pp = points + ((size_t)b * NPTS + idx) * 3;
    float* op = out + (size_t)gi * 3;
    op[0] = pp[0]; op[1] = pp[1]; op[2] = pp[2];
}

// ============================================================================
// Host launcher
// ============================================================================
extern "C" void kernel_launch(void* const* d_in, const int* in_sizes, int n_in,
                              void* d_out, int out_size, void* d_ws, size_t ws_size,
                              hipStream_t stream) {
    const float* points = (const float*)d_in[0];   // [B,N,3] f32
    const float* rois   = (const float*)d_in[1];   // [B,M,7] f32
    float* out = (float*)d_out;                    // [B,K,3] f32

    // ---- workspace carving (aligned) ----
    char* ws = (char*)d_ws;
    size_t o = 0;
    auto up = [](size_t x) { return (x + 255) & ~(size_t)255; };
    float4* centers4   = (float4*)(ws + o); o = up(o + (size_t)BATCH * NROI * sizeof(float4));
    float*  Tarr       = (float*) (ws + o); o = up(o + (size_t)BATCH * NROI * sizeof(float));
    int*    validCount = (int*)   (ws + o); o = up(o + (size_t)BATCH * sizeof(int));
    int*    svalid     = (int*)   (ws + o); o = up(o + (size_t)BATCH * NPTS * sizeof(int));
    float4* compactBuf = (float4*)(ws + o); o = up(o + (size_t)BATCH * NSEC * NPTS * sizeof(float4));
    int*    cnt        = (int*)   (ws + o); o = up(o + (size_t)BATCH * NSEC * sizeof(int));
    int*    nk         = (int*)   (ws + o); o = up(o + (size_t)BATCH * NSEC * sizeof(int));
    int*    offs       = (int*)   (ws + o); o = up(o + (size_t)BATCH * NSEC * sizeof(int));
    int*    stotal     = (int*)   (ws + o); o = up(o + (size_t)BATCH * sizeof(int));
    int*    fpsOut     = (int*)   (ws + o); o = up(o + (size_t)BATCH * NSEC * KKP * sizeof(int));

    spc_prep<<<1, 256, 0, stream>>>(rois, centers4, Tarr, validCount);

    spc_classify<<<BATCH * (NPTS / 128), 256, 0, stream>>>(
        points, centers4, Tarr, svalid, validCount);

    spc_compact<<<BATCH * NSEC, 1024, 0, stream>>>(points, svalid, compactBuf, cnt);

    spc_budget<<<1, 32, 0, stream>>>(validCount, cnt, nk, offs, stotal);

    const size_t fpsLds = 272 + (size_t)NPTS * sizeof(float);   // 262,416 B < 320KB
    spc_fps<<<BATCH * NSEC, 1024, fpsLds, stream>>>(compactBuf, cnt, nk, fpsOut);

    spc_assemble<<<(BATCH * KKP + 255) / 256, 256, 0, stream>>>(
        points, nk, offs, stotal, fpsOut, out);
}